// EnhancedFrameSampler_27333171871928
// MI455X (gfx1250) — compile-verified
//
#include <hip/hip_runtime.h>
#include <hip/hip_bf16.h>
#include <stdint.h>

// ---------------- problem constants (from reference) ----------------
#define NF   64
#define NH   384
#define NW   384
#define NPIX (NH * NW)                 // 147456 points per frame
#define NPTS ((size_t)NF * NPIX)       // 9,437,184 total points
#define RANK0 ((unsigned)(NPIX / 2 - 1))   // 73727  (0-indexed kth smallest)
#define RANK1 ((unsigned)(NPIX / 2))       // 73728
#define TS        131072u              // hash slots per frame (2^17, > max 73729 inserts)
#define EMPTYK    0xFFFFFFFFFFFFFFFFull
#define KEYB      2097152ll            // 1 << 21 (reference KEY_BASE)
#define CONF_THR  0.1f
#define TILE      1024                 // floats per staged LDS tile (4KB)
#define NT        (NPIX / TILE)        // 144 tiles per frame

// ---------------- CDNA5 async global->LDS path (guarded) ----------------
#if defined(__has_builtin)
#if __has_builtin(__builtin_amdgcn_global_load_async_to_lds_b128) && \
    __has_builtin(__builtin_amdgcn_s_wait_asynccnt)
#define HAVE_ASYNC_LDS 1
#endif
#endif

// Builtin signature (confirmed by hipcc diagnostic): AS1 v4i* src, AS3 v4i* dst, imm, imm.
typedef int v4i __attribute__((vector_size(4 * sizeof(int))));
typedef __attribute__((address_space(1))) v4i gv4i_t;   // global (__device__)
typedef __attribute__((address_space(3))) v4i lv4i_t;   // LDS (__shared__)

// 16B-per-lane async stage; falls back to a plain vector copy (then waits are no-ops).
__device__ __forceinline__ void stage16(const float* g, float* l) {
#if HAVE_ASYNC_LDS
  __builtin_amdgcn_global_load_async_to_lds_b128((gv4i_t*)g, (lv4i_t*)l, 0, 0);
#else
  *(float4*)l = *(const float4*)g;
#endif
}
__device__ __forceinline__ void wait_async_le1() {   // oldest copy done, newest may fly
#if HAVE_ASYNC_LDS
  __builtin_amdgcn_s_wait_asynccnt(1);
#endif
}
__device__ __forceinline__ void wait_async_le0() {   // all copies done
#if HAVE_ASYNC_LDS
  __builtin_amdgcn_s_wait_asynccnt(0);
#endif
}

// ---------------- ordered-uint encoding for float atomic min/max ----------------
__device__ __forceinline__ unsigned fenc(float v) {
  unsigned u = __float_as_uint(v);
  return (u & 0x80000000u) ? ~u : (u | 0x80000000u);
}
__device__ __forceinline__ float fdec(unsigned u) {
  return __uint_as_float((u & 0x80000000u) ? (u & 0x7FFFFFFFu) : ~u);
}

// ---------------- K0: init small workspace state ----------------
__global__ void init_kernel(unsigned* __restrict__ bbox, unsigned* __restrict__ nval,
                            double* __restrict__ dsums, unsigned* __restrict__ counts) {
  int t = threadIdx.x;
  if (t < 3) bbox[t] = 0xFFFFFFFFu;          // encoded +inf for min
  else if (t < 6) bbox[t] = 0u;              // encoded -inf for max
  if (t == 0) { *nval = 0u; dsums[0] = 0.0; dsums[1] = 0.0; }
  if (t < 256) counts[t] = 0u;               // [4][64] unique-voxel counters
}

// ---------------- K1: per-frame exact median via 12/12/8 radix select ----------------
// Confidences are positive floats -> bit pattern is order-isomorphic to uint.
// One workgroup per frame. Each pass streams the frame through double-buffered
// async LDS tiles: issue tile t+1, s_wait_asynccnt(1) -> tile t ready (in-order),
// histogram tile t while t+1 is in flight. Re-reads stay hot in the 192MB L2.
__global__ __launch_bounds__(256) void median_kernel(const float* __restrict__ conf,
                                                     float* __restrict__ qout) {
  __shared__ unsigned hist[4096];
  __shared__ unsigned s_bin, s_rank;
  __shared__ float buf[2][TILE];             // 2 x 4KB double buffer
  const int f = blockIdx.x;
  const float* cf = conf + (size_t)f * NPIX;
  const int tid = threadIdx.x;
  unsigned kth[2];
  for (int r = 0; r < 2; ++r) {
    unsigned target = (r == 0) ? RANK0 : RANK1;
    unsigned prefix = 0;
    for (int pass = 0; pass < 3; ++pass) {
      const unsigned nbins = (pass == 2) ? 256u : 4096u;
      for (int i = tid; i < 4096; i += 256) hist[i] = 0u;
      // prologue: stage tile 0
      stage16(cf + 0 * TILE + tid * 4, &buf[0][tid * 4]);
      for (int t = 0; t < NT; ++t) {
        if (t + 1 < NT) {
          stage16(cf + (size_t)(t + 1) * TILE + tid * 4, &buf[(t + 1) & 1][tid * 4]);
          wait_async_le1();                  // tile t complete, tile t+1 still flying
        } else {
          wait_async_le0();
        }
        __syncthreads();                     // tile t visible to all waves; hist zeroed (t==0)
        const float* cur = buf[t & 1];
        for (int i = tid; i < TILE; i += 256) {
          unsigned key = __float_as_uint(cur[i]);
          unsigned b; bool match;
          if (pass == 0)      { match = true;                  b = key >> 20; }
          else if (pass == 1) { match = (key >> 20) == prefix; b = (key >> 8) & 0xFFFu; }
          else                { match = (key >> 8)  == prefix; b = key & 0xFFu; }
          if (match) atomicAdd(&hist[b], 1u);
        }
        __syncthreads();                     // done reading buf[t&1] before it is re-staged
      }
      if (tid == 0) {
        unsigned acc = 0, b = 0;
        for (; b < nbins - 1; ++b) {
          unsigned c = hist[b];
          if (acc + c > target) break;
          acc += c;
        }
        s_bin = b; s_rank = target - acc;
      }
      __syncthreads();
      target = s_rank;
      if (pass == 0)      prefix = s_bin;
      else if (pass == 1) prefix = (prefix << 12) | s_bin;
      else                prefix = (prefix << 8)  | s_bin;
      __syncthreads();
    }
    kth[r] = prefix;
  }
  if (tid == 0)
    qout[f] = 0.5f * (__uint_as_float(kth[0]) + __uint_as_float(kth[1]));  // linear interp @ p=0.5
}

// ---------------- K2: streaming bbox + complexity moments ----------------
// Conf tile staged global->LDS with CDNA5 async-to-LDS; points loaded directly.
__global__ __launch_bounds__(256) void stats_kernel(
    const float* __restrict__ pts, const float* __restrict__ conf,
    const float* __restrict__ qv, unsigned* __restrict__ bbox,
    unsigned* __restrict__ nval, double* __restrict__ dsums) {
  __shared__ float lconf[TILE];
  __shared__ unsigned smn[3], smx[3], scnt;
  __shared__ double ssd, ssd2;
  const size_t base = (size_t)blockIdx.x * TILE;   // TILE | NPIX, chunk stays in one frame
  const int f = (int)(base / NPIX);
  const float qf = qv[f];
  if (threadIdx.x < 3) { smn[threadIdx.x] = 0xFFFFFFFFu; smx[threadIdx.x] = 0u; }
  if (threadIdx.x == 0) { scnt = 0u; ssd = 0.0; ssd2 = 0.0; }

  stage16(conf + base + (size_t)threadIdx.x * 4u, &lconf[threadIdx.x * 4u]);
  wait_async_le0();
  __syncthreads();

  float mn0 = 1e30f, mn1 = 1e30f, mn2 = 1e30f;
  float mx0 = -1e30f, mx1 = -1e30f, mx2 = -1e30f;
  float sd = 0.f, sd2 = 0.f;
  unsigned cnt = 0u;
  for (int i = threadIdx.x; i < TILE; i += 256) {
    float c = lconf[i];
    size_t gi = base + (size_t)i;
    float x = pts[3 * gi + 0], y = pts[3 * gi + 1], z = pts[3 * gi + 2];
    if (c > CONF_THR) {
      float d = sqrtf(x * x + y * y + z * z);
      sd += d; sd2 += d * d; ++cnt;
      if (c >= qf) {                                  // valid for bbox: conf>thr AND >= frame median
        mn0 = fminf(mn0, x); mx0 = fmaxf(mx0, x);
        mn1 = fminf(mn1, y); mx1 = fmaxf(mx1, y);
        mn2 = fminf(mn2, z); mx2 = fmaxf(mx2, z);
      }
    }
  }
  if (cnt) {
    atomicAdd(&ssd, (double)sd);
    atomicAdd(&ssd2, (double)sd2);
    atomicAdd(&scnt, cnt);
  }
  if (mn0 < 1e30f) {
    atomicMin(&smn[0], fenc(mn0)); atomicMax(&smx[0], fenc(mx0));
    atomicMin(&smn[1], fenc(mn1)); atomicMax(&smx[1], fenc(mx1));
    atomicMin(&smn[2], fenc(mn2)); atomicMax(&smx[2], fenc(mx2));
  }
  __syncthreads();
  if (threadIdx.x == 0 && scnt) {
    atomicAdd(nval, scnt);
    atomicAdd(dsums + 0, ssd);
    atomicAdd(dsums + 1, ssd2);
  }
  if (threadIdx.x < 3 && smn[threadIdx.x] != 0xFFFFFFFFu) {
    atomicMin(&bbox[threadIdx.x], smn[threadIdx.x]);
    atomicMax(&bbox[3 + threadIdx.x], smx[threadIdx.x]);
  }
}

// ---------------- K3: min_extent, voxel sizes, complexity scalar ----------------
__global__ void finalize_kernel(const unsigned* __restrict__ bbox,
                                const unsigned* __restrict__ nval,
                                const double* __restrict__ dsums,
                                float* __restrict__ pminf, float* __restrict__ vsize,
                                float* __restrict__ out_complexity) {
  if (threadIdx.x == 0) {
    float mn[3], mx[3];
    for (int k = 0; k < 3; ++k) { mn[k] = fdec(bbox[k]); mx[k] = fdec(bbox[3 + k]); }
    float ext = fminf(fminf(mx[0] - mn[0], mx[1] - mn[1]), mx[2] - mn[2]);
    pminf[0] = mn[0]; pminf[1] = mn[1]; pminf[2] = mn[2];
    const float lams[4] = {10.f, 20.f, 40.f, 20.f};   // VOXEL_SCALES + ADAPTIVE_LAMBDA
    for (int s = 0; s < 4; ++s) vsize[s] = ext / lams[s];
    double n = (double)(*nval);
    double sd = dsums[0], sd2 = dsums[1];
    double mean = sd / n;
    double var = (sd2 - n * mean * mean) / (n - 1.0);   // unbiased, moment form
    double dens = n / (double)NPTS;
    *out_complexity = (float)(sqrt(var) * dens);
  }
}

// ---------------- K4: reset hash tables (per scale) ----------------
__global__ __launch_bounds__(256) void clear_kernel(unsigned long long* __restrict__ tab) {
  size_t i = (size_t)blockIdx.x * 256 + threadIdx.x;
  tab[i] = EMPTYK;
}

// ---------------- K5: voxelize + hash-set insert, count unique per frame ----------------
__global__ __launch_bounds__(256) void insert_kernel(
    const float* __restrict__ pts, const float* __restrict__ conf,
    const float* __restrict__ qv, const float* __restrict__ pminf,
    const float* __restrict__ vsize, int s,
    unsigned long long* __restrict__ tables, unsigned* __restrict__ counts) {
  size_t gi = (size_t)blockIdx.x * 256 + threadIdx.x;
  int f = (int)(gi / NPIX);
  float c = conf[gi];
  float qf = qv[f];
  if (!(c > CONF_THR && c >= qf)) return;
  float vs = vsize[s];
  float x = pts[3 * gi + 0], y = pts[3 * gi + 1], z = pts[3 * gi + 2];
  long long cx = (long long)floorf((x - pminf[0]) / vs);
  long long cy = (long long)floorf((y - pminf[1]) / vs);
  long long cz = (long long)floorf((z - pminf[2]) / vs);
  unsigned long long key = (unsigned long long)((cx * KEYB + cy) * KEYB + cz);
  unsigned long long* tab = tables + (size_t)f * TS;
  unsigned h = (unsigned)((key * 0x9E3779B97F4A7C15ull) >> 40) & (TS - 1u);
  for (;;) {
    unsigned long long cur = tab[h];
    if (cur == key) return;                       // already counted
    if (cur == EMPTYK) {
      unsigned long long prev = atomicCAS(&tab[h], EMPTYK, key);
      if (prev == EMPTYK) { atomicAdd(&counts[s * NF + f], 1u); return; }
      if (prev == key) return;
    }
    h = (h + 1u) & (TS - 1u);
    __builtin_prefetch(&tab[h], 0, 0);            // -> global_prefetch_b8
  }
}

// ---------------- K6: counts -> float output ----------------
__global__ void writeout_kernel(const unsigned* __restrict__ counts, float* __restrict__ out) {
  int i = threadIdx.x;
  if (i < 256) out[i] = (float)counts[i];
}

// ---------------- host entry ----------------
extern "C" void kernel_launch(void* const* d_in, const int* in_sizes, int n_in,
                              void* d_out, int out_size, void* d_ws, size_t ws_size,
                              hipStream_t stream) {
  (void)in_sizes; (void)n_in; (void)out_size; (void)ws_size;
  const float* pts  = (const float*)d_in[0];   // [1,64,384,384,3]
  const float* conf = (const float*)d_in[1];   // [1,64,384,384]
  float* out = (float*)d_out;                  // [4*64] counts + [1] complexity

  char* ws = (char*)d_ws;
  float*    q      = (float*)(ws + 0);         // 64 frame medians
  unsigned* bbox   = (unsigned*)(ws + 256);    // 6 encoded min/max
  unsigned* nval   = (unsigned*)(ws + 288);
  double*   dsums  = (double*)(ws + 296);      // sum(d), sum(d^2)
  float*    pminf  = (float*)(ws + 320);
  float*    vsize  = (float*)(ws + 336);
  unsigned* counts = (unsigned*)(ws + 384);    // [4][64]
  unsigned long long* tables = (unsigned long long*)(ws + 4096);  // 64 * 2^17 * 8B = 64MB

  init_kernel<<<1, 256, 0, stream>>>(bbox, nval, dsums, counts);
  median_kernel<<<NF, 256, 0, stream>>>(conf, q);
  stats_kernel<<<(unsigned)(NPTS / TILE), 256, 0, stream>>>(pts, conf, q, bbox, nval, dsums);
  finalize_kernel<<<1, 32, 0, stream>>>(bbox, nval, dsums, pminf, vsize, out + 256);
  for (int s = 0; s < 4; ++s) {
    clear_kernel<<<(unsigned)(((size_t)NF * TS) / 256), 256, 0, stream>>>(tables);
    insert_kernel<<<(unsigned)(NPTS / 256), 256, 0, stream>>>(pts, conf, q, pminf, vsize, s,
                                                              tables, counts);
  }
  writeout_kernel<<<1, 256, 0, stream>>>(counts, out);
}